// HeterogeneousGNN_62113817034772
// MI455X (gfx1250) — compile-verified
//
#include <hip/hip_runtime.h>
#include <hip/hip_bf16.h>
#include <math.h>

// ---------------------------------------------------------------------------
// HeterogeneousGNN forward for MI455X (gfx1250, wave32, WMMA).
// All GEMMs run through V_WMMA_F32_16X16X4_F32 (full fp32 matrix pipe).
// Edge-bias einsums are algebraically factored through the 3-wide edge channel
// so the 134MB Eb tensor is never materialized (saves ~2.4GB HBM traffic).
// K-contiguous operands are loaded as 8-byte GLOBAL_LOAD_B64 via compile-time
// specialization (AK1/BK1) so each WMMA is fed by 2-3 wide loads, not 4 narrow.
// ---------------------------------------------------------------------------

typedef float v2f __attribute__((ext_vector_type(2)));
typedef float v8f __attribute__((ext_vector_type(8)));

#define B_SZ   4
#define U_SZ   128
#define T_SZ   512
#define E_DIM  128
#define H_DIM  4
#define D_HEAD 32
#define LN_EPS 1e-5f
#define INV_SQRT_D 0.17677669529663687f  // 1/sqrt(32)

__device__ __forceinline__ float hgnn_gelu(float x) {
  // exact erf GELU (torch nn.GELU default)
  return 0.5f * x * (1.0f + erff(x * 0.70710678118654752440f));
}

// ---------------------------------------------------------------------------
// Generic fp32 WMMA GEMM: C[m,n] = sum_k A[m,k]*B[k,n] (+bias[n]) (+GELU)
// A[m,k] = A[zoff + m*ldaM + k*ldaK], B[k,n] = B[zoff + k*ldbK + n*ldbN]
// Batched over blockIdx.z = b*nH + h with separate b/h strides.
// One wave computes a 16x16 tile; K stepped by 4.
// A frag (16x4 f32): lanes 0-15 row M=lane K={0,1}; lanes 16-31 same rows K={2,3}.
// B frag (4x16 f32): mirrored (lanes 0-15 col N=lane K={0,1}; hi lanes K={2,3}).
// C/D (16x16 f32, 8 VGPRs): vreg v -> (M = v + 8*(lane>=16), N = lane&15).
// AK1/BK1: operand is k-contiguous -> single b64 load per fragment.
// ---------------------------------------------------------------------------
template <bool AK1, bool BK1>
__global__ void hgnn_wmma_gemm(const float* __restrict__ A, const float* __restrict__ Bw,
                               float* __restrict__ C, const float* __restrict__ bias,
                               int M, int N, int K,
                               int ldaM, int ldaK, long long sAb, long long sAh,
                               int ldbK, int ldbN, long long sBb, long long sBh,
                               int ldc, long long sCb, long long sCh,
                               int nH, int act)
{
  int zb = blockIdx.z / nH;
  int zh = blockIdx.z % nH;
  const float* Ab = A + zb * sAb + zh * sAh;
  const float* Bb = Bw + zb * sBb + zh * sBh;
  float* Cb = C + zb * sCb + zh * sCh;

  int tilesN = N >> 4;
  int tile = blockIdx.x * (blockDim.x >> 5) + (threadIdx.x >> 5);
  if (tile >= (M >> 4) * tilesN) return;  // uniform per wave -> EXEC stays all-1s

  int tm = (tile / tilesN) << 4;
  int tn = (tile % tilesN) << 4;
  int lane = threadIdx.x & 31;
  int r = lane & 15;
  int kk = (lane >> 4) << 1;  // 0 for lanes 0-15, 2 for lanes 16-31

  const float* Ar = Ab + (long long)(tm + r) * ldaM;
  const float* Bc = Bb + (long long)(tn + r) * ldbN;

  v8f acc = {};
  for (int k0 = 0; k0 < K; k0 += 4) {
    v2f a, b;
    if (AK1) {
      a = *(const v2f*)(Ar + k0 + kk);          // GLOBAL_LOAD_B64 (8B aligned)
    } else {
      a.x = Ar[(k0 + kk) * ldaK];
      a.y = Ar[(k0 + kk + 1) * ldaK];
    }
    if (BK1) {
      b = *(const v2f*)(Bc + k0 + kk);          // GLOBAL_LOAD_B64 (8B aligned)
    } else {
      b.x = Bc[(k0 + kk) * ldbK];
      b.y = Bc[(k0 + kk + 1) * ldbK];
    }
    acc = __builtin_amdgcn_wmma_f32_16x16x4_f32(false, a, false, b, (short)0, acc,
                                                false, false);
  }

  int rbase = tm + ((lane >> 4) << 3);
  int col = tn + r;
  float bv = bias ? bias[col] : 0.0f;
#pragma unroll
  for (int v = 0; v < 8; ++v) {
    float val = acc[v] + bv;
    if (act == 1) val = hgnn_gelu(val);
    Cb[(long long)(rbase + v) * ldc + col] = val;
  }
}

// ---------------------------------------------------------------------------
// Naive dense: out[m,n] = act(sum_k in[m,k]*w[k,n] + bias[n]). For tiny/odd
// shapes (K=3/4 input proj, 4-row graph head) where WMMA shapes don't apply.
// ---------------------------------------------------------------------------
__global__ void hgnn_dense(const float* __restrict__ in, const float* __restrict__ w,
                           const float* __restrict__ bias, float* __restrict__ out,
                           int M, int N, int K, int act)
{
  int idx = blockIdx.x * blockDim.x + threadIdx.x;
  if (idx >= M * N) return;
  int m = idx / N, n = idx % N;
  float acc = bias ? bias[n] : 0.0f;
  const float* ir = in + (long long)m * K;
  for (int k = 0; k < K; ++k) acc += ir[k] * w[k * N + n];
  if (act == 1) acc = hgnn_gelu(acc);
  out[idx] = acc;
}

// ---------------------------------------------------------------------------
// qe[b,h,n,c] = sum_d Q[b,n,h*32+d] * ep_w[c, h*32+d]   (c=0..2)
// qe[b,h,n,3] = sum_d Q[b,n,h*32+d] * ep_b[h*32+d]
// ---------------------------------------------------------------------------
__global__ void hgnn_qe(const float* __restrict__ Q, const float* __restrict__ ep_w,
                        const float* __restrict__ ep_b, float* __restrict__ qe, int Nq)
{
  int idx = blockIdx.x * blockDim.x + threadIdx.x;
  if (idx >= B_SZ * H_DIM * Nq) return;
  int n = idx % Nq;
  int bh = idx / Nq;
  int h = bh % H_DIM;
  int b = bh / H_DIM;
  const float* qr = Q + ((long long)(b * Nq + n)) * E_DIM + h * D_HEAD;
  const float* w0 = ep_w + h * D_HEAD;
  const float* bb = ep_b + h * D_HEAD;
  float c0 = 0.f, c1 = 0.f, c2 = 0.f, cb = 0.f;
#pragma unroll 4
  for (int d = 0; d < D_HEAD; ++d) {
    float q = qr[d];
    c0 += q * w0[d];
    c1 += q * w0[E_DIM + d];
    c2 += q * w0[2 * E_DIM + d];
    cb += q * bb[d];
  }
  float* o = qe + (long long)idx * 4;
  o[0] = c0; o[1] = c1; o[2] = c2; o[3] = cb;
}

// ---------------------------------------------------------------------------
// Fused: scores -> add edge-bias term -> scale -> softmax -> attn (in place),
// and accumulate ae[b,h,n,c] = sum_m attn * edge[...,c].
// One 128-thread block per (b,h,n) row; M = Nk in {128, 512}.
// edge addr = b*(U*T*3) + n*eN + m*eM + c  (eN/eM encode the transpose).
// ---------------------------------------------------------------------------
__global__ void hgnn_softmax_edge(float* __restrict__ scores, const float* __restrict__ edge,
                                  const float* __restrict__ qe, float* __restrict__ ae,
                                  int Nq, int M, long long eN, long long eM)
{
  __shared__ float sm[128];
  int row = blockIdx.x;
  int tid = threadIdx.x;
  int n = row % Nq;
  int bh = row / Nq;
  int b = bh / H_DIM;
  long long base = (long long)row * M;
  const float* q4 = qe + (long long)row * 4;
  float q0 = q4[0], q1 = q4[1], q2 = q4[2], qb = q4[3];
  const float* eb = edge + (long long)b * (U_SZ * T_SZ * 3) + (long long)n * eN;

  int nel = M >> 7;  // 1 or 4
  float sv[4], e0v[4], e1v[4], e2v[4];
  float mx = -3.4e38f;
  for (int j = 0; j < nel; ++j) {
    int m = tid + (j << 7);
    const float* ep = eb + (long long)m * eM;
    float e0 = ep[0], e1 = ep[1], e2 = ep[2];
    float s = (scores[base + m] + e0 * q0 + e1 * q1 + e2 * q2 + qb) * INV_SQRT_D;
    sv[j] = s; e0v[j] = e0; e1v[j] = e1; e2v[j] = e2;
    mx = fmaxf(mx, s);
  }
  sm[tid] = mx; __syncthreads();
  for (int s = 64; s > 0; s >>= 1) { if (tid < s) sm[tid] = fmaxf(sm[tid], sm[tid + s]); __syncthreads(); }
  mx = sm[0]; __syncthreads();

  float sum = 0.f;
  for (int j = 0; j < nel; ++j) { sv[j] = expf(sv[j] - mx); sum += sv[j]; }
  sm[tid] = sum; __syncthreads();
  for (int s = 64; s > 0; s >>= 1) { if (tid < s) sm[tid] += sm[tid + s]; __syncthreads(); }
  float inv = 1.0f / sm[0]; __syncthreads();

  float a0 = 0.f, a1 = 0.f, a2 = 0.f;
  for (int j = 0; j < nel; ++j) {
    int m = tid + (j << 7);
    float w = sv[j] * inv;
    scores[base + m] = w;
    a0 += w * e0v[j]; a1 += w * e1v[j]; a2 += w * e2v[j];
  }
  sm[tid] = a0; __syncthreads();
  for (int s = 64; s > 0; s >>= 1) { if (tid < s) sm[tid] += sm[tid + s]; __syncthreads(); }
  if (tid == 0) ae[(long long)row * 4 + 0] = sm[0];
  __syncthreads();
  sm[tid] = a1; __syncthreads();
  for (int s = 64; s > 0; s >>= 1) { if (tid < s) sm[tid] += sm[tid + s]; __syncthreads(); }
  if (tid == 0) ae[(long long)row * 4 + 1] = sm[0];
  __syncthreads();
  sm[tid] = a2; __syncthreads();
  for (int s = 64; s > 0; s >>= 1) { if (tid < s) sm[tid] += sm[tid + s]; __syncthreads(); }
  if (tid == 0) ae[(long long)row * 4 + 2] = sm[0];
}

// ctx[b,n,e] += ae[b,h(e),n,:] . ep_w[:,e] + ep_b[e]   (sum_m attn = 1)
__global__ void hgnn_add_edge_ctx(float* __restrict__ ctx, const float* __restrict__ ae,
                                  const float* __restrict__ ep_w, const float* __restrict__ ep_b,
                                  int Nq)
{
  int idx = blockIdx.x * blockDim.x + threadIdx.x;
  if (idx >= B_SZ * Nq * E_DIM) return;
  int e = idx & (E_DIM - 1);
  int rn = idx >> 7;
  int n = rn % Nq;
  int b = rn / Nq;
  int h = e >> 5;
  const float* a = ae + ((long long)((b * H_DIM + h) * Nq + n)) * 4;
  ctx[idx] += a[0] * ep_w[e] + a[1] * ep_w[E_DIM + e] + a[2] * ep_w[2 * E_DIM + e] + ep_b[e];
}

// LayerNorm over E=128: out[row*ldo+e] = LN(in1 (+in2)) (*g +b). g==null -> plain LN.
__global__ void hgnn_ln(const float* __restrict__ in1, const float* __restrict__ in2,
                        const float* __restrict__ g, const float* __restrict__ bta,
                        float* __restrict__ out, int ldo)
{
  __shared__ float sm[128];
  int row = blockIdx.x;
  int e = threadIdx.x;
  float x = in1[(long long)row * E_DIM + e];
  if (in2) x += in2[(long long)row * E_DIM + e];
  sm[e] = x; __syncthreads();
  for (int s = 64; s > 0; s >>= 1) { if (e < s) sm[e] += sm[e + s]; __syncthreads(); }
  float mean = sm[0] * (1.0f / E_DIM); __syncthreads();
  float d = x - mean;
  sm[e] = d * d; __syncthreads();
  for (int s = 64; s > 0; s >>= 1) { if (e < s) sm[e] += sm[e + s]; __syncthreads(); }
  float var = sm[0] * (1.0f / E_DIM);
  float y = d * rsqrtf(var + LN_EPS);
  if (g) y = y * g[e] + bta[e];
  out[(long long)row * ldo + e] = y;
}

// s[row] = tanh(x_row @ w1 + b1) @ w2 + b2 ; 64 threads per row.
__global__ void hgnn_pool_score(const float* __restrict__ x, const float* __restrict__ w1,
                                const float* __restrict__ b1, const float* __restrict__ w2,
                                const float* __restrict__ b2, float* __restrict__ s)
{
  __shared__ float sm[64];
  int row = blockIdx.x;
  int j = threadIdx.x;
  const float* xr = x + (long long)row * E_DIM;
  float acc = b1[j];
  for (int e = 0; e < E_DIM; ++e) acc += xr[e] * w1[e * 64 + j];
  sm[j] = tanhf(acc) * w2[j];
  __syncthreads();
  for (int st = 32; st > 0; st >>= 1) { if (j < st) sm[j] += sm[j + st]; __syncthreads(); }
  if (j == 0) s[row] = sm[0] + b2[0];
}

// pooled[b,e] = sum_n softmax_n(s[b,:]) * x[b,n,e]  (pre-LN)
__global__ void hgnn_pool_finalize(const float* __restrict__ x, const float* __restrict__ s,
                                   float* __restrict__ pooled, int Nn)
{
  __shared__ float sm[128];
  __shared__ float stats[2];
  int b = blockIdx.x;
  int tid = threadIdx.x;
  const float* sb = s + (long long)b * Nn;
  float mx = -3.4e38f;
  for (int n = tid; n < Nn; n += 128) mx = fmaxf(mx, sb[n]);
  sm[tid] = mx; __syncthreads();
  for (int st = 64; st > 0; st >>= 1) { if (tid < st) sm[tid] = fmaxf(sm[tid], sm[tid + st]); __syncthreads(); }
  if (tid == 0) stats[0] = sm[0];
  __syncthreads();
  mx = stats[0]; __syncthreads();
  float sum = 0.f;
  for (int n = tid; n < Nn; n += 128) sum += expf(sb[n] - mx);
  sm[tid] = sum; __syncthreads();
  for (int st = 64; st > 0; st >>= 1) { if (tid < st) sm[tid] += sm[tid + st]; __syncthreads(); }
  if (tid == 0) stats[1] = 1.0f / sm[0];
  __syncthreads();
  float inv = stats[1];
  const float* xb = x + (long long)b * Nn * E_DIM;
  float acc = 0.f;
  for (int n = 0; n < Nn; ++n) acc += expf(sb[n] - mx) * inv * xb[(long long)n * E_DIM + tid];
  pooled[b * E_DIM + tid] = acc;
}

// ---------------------------------------------------------------------------
// Host side
// ---------------------------------------------------------------------------
struct AttnP { const float *ep_b, *ep_w, *ln_b, *ln_g, *wk, *wo_b, *wo_w, *wq, *wv; };
struct MlpP  { const float *b1, *b2, *ln_b, *ln_g, *w1, *w2; };
struct PoolP { const float *b1, *b2, *ln_b, *ln_g, *w1, *w2; };

extern "C" void kernel_launch(void* const* d_in, const int* in_sizes, int n_in,
                              void* d_out, int out_size, void* d_ws, size_t ws_size,
                              hipStream_t stream)
{
  (void)in_sizes; (void)n_in; (void)out_size; (void)ws_size;
  const float* usvF  = (const float*)d_in[0];   // [B,U,3]
  const float* taskF = (const float*)d_in[1];   // [B,T,4]
  const float* edges = (const float*)d_in[2];   // [B,U,T,3]

  // params flattened as jax pytree leaves (dict keys sorted, recursive)
  auto F = [&](int i) { return (const float*)d_in[i]; };
  int pi = 3;
  const float* gn_b  = F(pi++); const float* gn_g  = F(pi++);
  const float* gp_b1 = F(pi++); const float* gp_b2 = F(pi++);
  const float* gp_w1 = F(pi++); const float* gp_w2 = F(pi++);
  AttnP t2u[3], u2t[3]; MlpP tffn[3], uffn[3];
  for (int l = 0; l < 3; ++l) {
    AttnP* a; MlpP* m;
    a = &t2u[l];
    a->ep_b = F(pi++); a->ep_w = F(pi++); a->ln_b = F(pi++); a->ln_g = F(pi++);
    a->wk = F(pi++); a->wo_b = F(pi++); a->wo_w = F(pi++); a->wq = F(pi++); a->wv = F(pi++);
    m = &tffn[l];
    m->b1 = F(pi++); m->b2 = F(pi++); m->ln_b = F(pi++); m->ln_g = F(pi++);
    m->w1 = F(pi++); m->w2 = F(pi++);
    a = &u2t[l];
    a->ep_b = F(pi++); a->ep_w = F(pi++); a->ln_b = F(pi++); a->ln_g = F(pi++);
    a->wk = F(pi++); a->wo_b = F(pi++); a->wo_w = F(pi++); a->wq = F(pi++); a->wv = F(pi++);
    m = &uffn[l];
    m->b1 = F(pi++); m->b2 = F(pi++); m->ln_b = F(pi++); m->ln_g = F(pi++);
    m->w1 = F(pi++); m->w2 = F(pi++);
  }
  PoolP pool_t, pool_u;
  pool_t.b1 = F(pi++); pool_t.b2 = F(pi++); pool_t.ln_b = F(pi++); pool_t.ln_g = F(pi++);
  pool_t.w1 = F(pi++); pool_t.w2 = F(pi++);
  pool_u.b1 = F(pi++); pool_u.b2 = F(pi++); pool_u.ln_b = F(pi++); pool_u.ln_g = F(pi++);
  pool_u.w1 = F(pi++); pool_u.w2 = F(pi++);
  const float* task_b = F(pi++); const float* task_w = F(pi++);
  const float* usv_b  = F(pi++); const float* usv_w  = F(pi++);

  // workspace layout (floats); total ~15.5 MB
  float* p = (float*)d_ws;
  float* usvA  = p; p += B_SZ * U_SZ * E_DIM;   // 65536
  float* usvB  = p; p += B_SZ * U_SZ * E_DIM;
  float* taskA = p; p += B_SZ * T_SZ * E_DIM;   // 262144
  float* taskB = p; p += B_SZ * T_SZ * E_DIM;
  float* Qb    = p; p += B_SZ * T_SZ * E_DIM;
  float* Kb    = p; p += B_SZ * T_SZ * E_DIM;
  float* Vb    = p; p += B_SZ * T_SZ * E_DIM;
  float* Sc    = p; p += B_SZ * H_DIM * U_SZ * T_SZ;  // 1048576
  float* QE    = p; p += B_SZ * H_DIM * T_SZ * 4;
  float* AE    = p; p += B_SZ * H_DIM * T_SZ * 4;
  float* Ctx   = p; p += B_SZ * T_SZ * E_DIM;
  float* Xat   = p; p += B_SZ * T_SZ * E_DIM;
  float* Hid   = p; p += B_SZ * T_SZ * 2 * E_DIM;     // 524288
  float* Tmp   = p; p += B_SZ * T_SZ * E_DIM;
  float* SbU   = p; p += B_SZ * T_SZ;
  float* SbT   = p; p += B_SZ * T_SZ;
  float* PoU   = p; p += B_SZ * E_DIM;
  float* PoT   = p; p += B_SZ * E_DIM;
  float* Gc    = p; p += B_SZ * 2 * E_DIM;
  float* G1    = p; p += B_SZ * E_DIM;
  float* G2    = p; p += B_SZ * E_DIM;

  float* outU = (float*)d_out;
  float* outT = outU + B_SZ * U_SZ * E_DIM;
  float* outG = outT + B_SZ * T_SZ * E_DIM;

  auto gemm = [&](const float* A, const float* Bw, float* C, const float* bias,
                  int M, int N, int K,
                  int ldaM, int ldaK, long long sAb, long long sAh,
                  int ldbK, int ldbN, long long sBb, long long sBh,
                  int ldc, long long sCb, long long sCh,
                  int zB, int zH, int act) {
    int tiles = (M / 16) * (N / 16);
    dim3 grid((tiles + 3) / 4, 1, zB * zH);
    bool ak1 = (ldaK == 1);
    bool bk1 = (ldbK == 1);
    if (ak1 && bk1)
      hgnn_wmma_gemm<true, true><<<grid, 128, 0, stream>>>(A, Bw, C, bias, M, N, K,
          ldaM, ldaK, sAb, sAh, ldbK, ldbN, sBb, sBh, ldc, sCb, sCh, zH, act);
    else if (ak1)
      hgnn_wmma_gemm<true, false><<<grid, 128, 0, stream>>>(A, Bw, C, bias, M, N, K,
          ldaM, ldaK, sAb, sAh, ldbK, ldbN, sBb, sBh, ldc, sCb, sCh, zH, act);
    else if (bk1)
      hgnn_wmma_gemm<false, true><<<grid, 128, 0, stream>>>(A, Bw, C, bias, M, N, K,
          ldaM, ldaK, sAb, sAh, ldbK, ldbN, sBb, sBh, ldc, sCb, sCh, zH, act);
    else
      hgnn_wmma_gemm<false, false><<<grid, 128, 0, stream>>>(A, Bw, C, bias, M, N, K,
          ldaM, ldaK, sAb, sAh, ldbK, ldbN, sBb, sBh, ldc, sCb, sCh, zH, act);
  };

  auto run_attn = [&](const AttnP& ap, const float* qin, int Nq, const float* kvin, int Nk,
                      long long eN, long long eM, float* xout) {
    // projections (no bias on wq/wk/wv in reference)
    gemm(qin,  ap.wq, Qb, nullptr, B_SZ * Nq, E_DIM, E_DIM,
         E_DIM, 1, 0, 0,  E_DIM, 1, 0, 0,  E_DIM, 0, 0, 1, 1, 0);
    gemm(kvin, ap.wk, Kb, nullptr, B_SZ * Nk, E_DIM, E_DIM,
         E_DIM, 1, 0, 0,  E_DIM, 1, 0, 0,  E_DIM, 0, 0, 1, 1, 0);
    gemm(kvin, ap.wv, Vb, nullptr, B_SZ * Nk, E_DIM, E_DIM,
         E_DIM, 1, 0, 0,  E_DIM, 1, 0, 0,  E_DIM, 0, 0, 1, 1, 0);
    // edge-bias dot factors
    int nqe = B_SZ * H_DIM * Nq;
    hgnn_qe<<<(nqe + 255) / 256, 256, 0, stream>>>(Qb, ap.ep_w, ap.ep_b, QE, Nq);
    // scores = Q_h @ K_h^T  (batched over b,h); both operands k-contiguous
    gemm(Qb, Kb, Sc, nullptr, Nq, Nk, D_HEAD,
         E_DIM, 1, (long long)Nq * E_DIM, D_HEAD,
         1, E_DIM, (long long)Nk * E_DIM, D_HEAD,
         Nk, (long long)H_DIM * Nq * Nk, (long long)Nq * Nk,
         B_SZ, H_DIM, 0);
    // fused edge-bias + scale + softmax + ae accumulation
    hgnn_softmax_edge<<<B_SZ * H_DIM * Nq, 128, 0, stream>>>(Sc, edges, QE, AE, Nq, Nk, eN, eM);
    // ctx = attn @ V_h
    gemm(Sc, Vb, Ctx, nullptr, Nq, D_HEAD, Nk,
         Nk, 1, (long long)H_DIM * Nq * Nk, (long long)Nq * Nk,
         E_DIM, 1, (long long)Nk * E_DIM, D_HEAD,
         E_DIM, (long long)Nq * E_DIM, D_HEAD,
         B_SZ, H_DIM, 0);
    int nel = B_SZ * Nq * E_DIM;
    hgnn_add_edge_ctx<<<(nel + 255) / 256, 256, 0, stream>>>(Ctx, AE, ap.ep_w, ap.ep_b, Nq);
    // output projection + residual LN
    gemm(Ctx, ap.wo_w, Tmp, ap.wo_b, B_SZ * Nq, E_DIM, E_DIM,
         E_DIM, 1, 0, 0,  E_DIM, 1, 0, 0,  E_DIM, 0, 0, 1, 1, 0);
    hgnn_ln<<<B_SZ * Nq, 128, 0, stream>>>(qin, Tmp, ap.ln_g, ap.ln_b, xout, E_DIM);
  };

  auto run_mlp = [&](const MlpP& mp, const float* xin, int Nq, float* xout) {
    gemm(xin, mp.w1, Hid, mp.b1, B_SZ * Nq, 2 * E_DIM, E_DIM,
         E_DIM, 1, 0, 0,  2 * E_DIM, 1, 0, 0,  2 * E_DIM, 0, 0, 1, 1, 1 /*gelu*/);
    gemm(Hid, mp.w2, Tmp, mp.b2, B_SZ * Nq, E_DIM, 2 * E_DIM,
         2 * E_DIM, 1, 0, 0,  E_DIM, 1, 0, 0,  E_DIM, 0, 0, 1, 1, 0);
    hgnn_ln<<<B_SZ * Nq, 128, 0, stream>>>(xin, Tmp, mp.ln_g, mp.ln_b, xout, E_DIM);
  };

  // input projections (K=3/4, too small for WMMA)
  hgnn_dense<<<(B_SZ * U_SZ * E_DIM + 255) / 256, 256, 0, stream>>>(usvF, usv_w, usv_b, usvA, B_SZ * U_SZ, E_DIM, 3, 0);
  hgnn_dense<<<(B_SZ * T_SZ * E_DIM + 255) / 256, 256, 0, stream>>>(taskF, task_w, task_b, taskA, B_SZ * T_SZ, E_DIM, 4, 0);

  const float* ucur = usvA;
  const float* tcur = taskA;
  for (int l = 0; l < 3; ++l) {
    float* uo = (l == 2) ? outU : ((l & 1) ? usvA : usvB);
    float* to = (l == 2) ? outT : ((l & 1) ? taskA : taskB);
    // usv <- cross_attn(u2t, usv, task, task, edge) ; edge[b, n=u, m=t]
    run_attn(u2t[l], ucur, U_SZ, tcur, T_SZ, (long long)T_SZ * 3, 3LL, Xat);
    run_mlp(uffn[l], Xat, U_SZ, uo);
    // task <- cross_attn(t2u, task, usv, usv, edge^T) ; edge[b, m=u, n=t]
    run_attn(t2u[l], tcur, T_SZ, ucur, U_SZ, 3LL, (long long)T_SZ * 3, Xat);
    run_mlp(tffn[l], Xat, T_SZ, to);
    if (l < 2) {  // inter-layer parameter-free LN
      hgnn_ln<<<B_SZ * U_SZ, 128, 0, stream>>>(uo, nullptr, nullptr, nullptr, uo, E_DIM);
      hgnn_ln<<<B_SZ * T_SZ, 128, 0, stream>>>(to, nullptr, nullptr, nullptr, to, E_DIM);
    }
    ucur = uo; tcur = to;
  }

  // attention pooling -> concat into Gc[b, 0:128]=pool_u, [128:256]=pool_t
  hgnn_pool_score<<<B_SZ * U_SZ, 64, 0, stream>>>(ucur, pool_u.w1, pool_u.b1, pool_u.w2, pool_u.b2, SbU);
  hgnn_pool_finalize<<<B_SZ, 128, 0, stream>>>(ucur, SbU, PoU, U_SZ);
  hgnn_ln<<<B_SZ, 128, 0, stream>>>(PoU, nullptr, pool_u.ln_g, pool_u.ln_b, Gc, 2 * E_DIM);
  hgnn_pool_score<<<B_SZ * T_SZ, 64, 0, stream>>>(tcur, pool_t.w1, pool_t.b1, pool_t.w2, pool_t.b2, SbT);
  hgnn_pool_finalize<<<B_SZ, 128, 0, stream>>>(tcur, SbT, PoT, T_SZ);
  hgnn_ln<<<B_SZ, 128, 0, stream>>>(PoT, nullptr, pool_t.ln_g, pool_t.ln_b, Gc + E_DIM, 2 * E_DIM);

  // graph head (M=4 rows: naive dense)
  hgnn_dense<<<(B_SZ * E_DIM + 255) / 256, 256, 0, stream>>>(Gc, gp_w1, gp_b1, G1, B_SZ, E_DIM, 2 * E_DIM, 1);
  hgnn_dense<<<(B_SZ * E_DIM + 255) / 256, 256, 0, stream>>>(G1, gp_w2, gp_b2, G2, B_SZ, E_DIM, E_DIM, 0);
  hgnn_ln<<<B_SZ, 128, 0, stream>>>(G2, nullptr, gn_g, gn_b, outG, E_DIM);
}